// cross_scale_trans_68539088109998
// MI455X (gfx1250) — compile-verified
//
#include <hip/hip_runtime.h>
#include <hip/hip_bf16.h>

#define NPTS 16384
#define MNB 16
#define NHEADS 8
#define DIM 256
#define CIN 64
#define DFFNV 512
#define DHEAD 32
#define GRIDSZ 128
#define EPSV 1e-5f

typedef __attribute__((ext_vector_type(2))) float v2f;
typedef __attribute__((ext_vector_type(8))) float v8f;

__device__ __forceinline__ int iabs_(int v) { return v < 0 ? -v : v; }

// A-matrix 16x4 fp32 fragment from LDS (row-major, stride in floats).
// lanes 0-15: rows 0-15, K pair {k0,k0+1}; lanes 16-31: rows 0-15, K pair {k0+2,k0+3}.
__device__ __forceinline__ v2f lds_a4(const float* A, int stride, int k0) {
  int lane = threadIdx.x & 31;
  int row = lane & 15;
  int kk = k0 + ((lane >> 4) << 1);
  v2f a;
  a.x = A[row * stride + kk];
  a.y = A[row * stride + kk + 1];
  return a;
}

// C += A(16xK from LDS) * B-tile from fragment-major packed weights.
// Packed layout: float2 per lane per fragment, index ((jt*Kq + kq)*32 + lane).
// One coalesced global_load_b64 per WMMA instead of two strided b32 loads.
__device__ __forceinline__ v8f gemm_tile_p(const float* A, int astride,
                                           const float* Bp, int Kq, int jt,
                                           v8f c) {
  const float2* bp =
      (const float2*)Bp + (size_t)jt * Kq * 32 + (threadIdx.x & 31);
  // speculative prefetch of the next tile's fragments (dropped if OOB)
  __builtin_prefetch((const void*)(bp + (size_t)Kq * 32), 0, 0);
#pragma unroll 4
  for (int kq = 0; kq < Kq; ++kq) {
    v2f a = lds_a4(A, astride, kq * 4);
    float2 bf = bp[(size_t)kq * 32];
    v2f b;
    b.x = bf.x;
    b.y = bf.y;
    c = __builtin_amdgcn_wmma_f32_16x16x4_f32(false, a, false, b, (short)0, c,
                                              false, false);
  }
  return c;
}

// ------------- Weight repack: row-major (K,OUT) -> fragment-major -------------
// Bp[((jt*(K/4)+kq)*32 + lane)*2 + which] = W[kq*4 + (lane>>4)*2 + which][jt*16 + (lane&15)]
__global__ void pack_b_kernel(const float* __restrict__ W, int K, int OUT,
                              float* __restrict__ Bp) {
  int total = K * OUT;
  int i = blockIdx.x * blockDim.x + threadIdx.x;
  if (i >= total) return;
  int which = i & 1;
  int lane = (i >> 1) & 31;
  int kq = (i >> 6) % (K / 4);
  int jt = (i >> 6) / (K / 4);
  int row = kq * 4 + ((lane >> 4) << 1) + which;
  int col = jt * 16 + (lane & 15);
  Bp[i] = W[(size_t)row * OUT + col];
}

// ---------------- Stage A1: positional-encoding hidden layer ----------------
__global__ void pe_hidden_kernel(const int* __restrict__ coords,
                                 const float* __restrict__ pw1,
                                 const float* __restrict__ pb1,
                                 float* __restrict__ hid) {
  int i = blockIdx.x * blockDim.x + threadIdx.x;
  if (i >= NPTS * 128) return;
  int n = i >> 7, d = i & 127;
  float inv = 1.0f / (float)(GRIDSZ - 1);
  float x0 = (float)coords[n * 3 + 0] * inv;
  float x1 = (float)coords[n * 3 + 1] * inv;
  float x2 = (float)coords[n * 3 + 2] * inv;
  float h = x0 * pw1[d] + x1 * pw1[128 + d] + x2 * pw1[256 + d] + pb1[d];
  hid[i] = fmaxf(h, 0.0f);
}

// ---------------- Stage B: Manhattan top-16 neighbor search ----------------
__global__ __launch_bounds__(256) void neighbor_kernel(
    const int* __restrict__ coords, int* __restrict__ nbr) {
  __shared__ int pc[NPTS];
  const int tid = threadIdx.x;
  for (int t = tid; t < NPTS; t += 256)
    pc[t] = coords[t * 3 + 0] | (coords[t * 3 + 1] << 8) |
            (coords[t * 3 + 2] << 16);
  __syncthreads();
  const int n = blockIdx.x * 256 + tid;
  const int q = pc[n];
  const int qx = q & 255, qy = (q >> 8) & 255, qz = (q >> 16) & 255;
  int c0 = 0, c1 = 0, c2 = 0;
  for (int j = 0; j < NPTS; ++j) {
    int p = pc[j];
    int d = iabs_((p & 255) - qx) + iabs_(((p >> 8) & 255) - qy) +
            iabs_(((p >> 16) & 255) - qz);
    c0 += (d == 0);
    c1 += (d == 1);
    c2 += (d == 2);
  }
  int r0 = 0, r1 = 0, r2 = 0;
  for (int j = 0; j < NPTS; ++j) {
    int p = pc[j];
    int d = iabs_((p & 255) - qx) + iabs_(((p >> 8) & 255) - qy) +
            iabs_(((p >> 16) & 255) - qz);
    if (d == 0) {
      if (r0 < MNB) nbr[n * MNB + r0] = j;
      ++r0;
    } else if (d == 1) {
      int pos = c0 + r1;
      if (pos < MNB) nbr[n * MNB + pos] = j;
      ++r1;
    } else if (d == 2) {
      int pos = c0 + c1 + r2;
      if (pos < MNB) nbr[n * MNB + pos] = j;
      ++r2;
    }
  }
  int total = c0 + c1 + c2;
  if (total > MNB) total = MNB;
  for (int pos = total; pos < MNB; ++pos) nbr[n * MNB + pos] = -1;
}

// ---------------- Stage A2: src = features@proj_w + hidden@pe_w2 + biases ---
__global__ __launch_bounds__(256) void src_kernel(
    const float* __restrict__ feats, const float* __restrict__ hid,
    const float* __restrict__ p_projw, const float* __restrict__ projb,
    const float* __restrict__ p_pew2, const float* __restrict__ peb2,
    float* __restrict__ srcg) {
  __shared__ __align__(16) float Ft[16 * 68];
  __shared__ __align__(16) float Ht[16 * 132];
  const int tid = threadIdx.x;
  const int w = tid >> 5;
  const int lane = tid & 31;
  const int n0 = blockIdx.x * 16;
  for (int t = tid; t < 16 * 16; t += 256) {
    int row = t >> 4, qq = t & 15;
    ((float4*)&Ft[row * 68])[qq] =
        ((const float4*)(feats + (size_t)(n0 + row) * CIN))[qq];
  }
  for (int t = tid; t < 16 * 32; t += 256) {
    int row = t >> 5, qq = t & 31;
    ((float4*)&Ht[row * 132])[qq] =
        ((const float4*)(hid + (size_t)(n0 + row) * 128))[qq];
  }
  __syncthreads();
  for (int jj = 0; jj < 2; ++jj) {
    int jt = w * 2 + jj;
    v8f c = {};
    c = gemm_tile_p(Ft, 68, p_projw, CIN / 4, jt, c);
    c = gemm_tile_p(Ht, 132, p_pew2, 128 / 4, jt, c);
    int col = lane & 15, half = lane >> 4;
#pragma unroll
    for (int i = 0; i < 8; ++i) {
      int r = half * 8 + i, d = jt * 16 + col;
      srcg[(size_t)(n0 + r) * DIM + d] = c[i] + projb[d] + peb2[d];
    }
  }
}

// ---------------- Stage C: fused gather + QKV + attention ----------------
__global__ __launch_bounds__(128) void attn_kernel(
    const float* __restrict__ src, const int* __restrict__ nbr,
    const float* __restrict__ p_wq, const float* __restrict__ bq,
    const float* __restrict__ p_wk, const float* __restrict__ bk,
    const float* __restrict__ p_wv, const float* __restrict__ bv,
    float* __restrict__ aog) {
  __shared__ __align__(16) float Anf[4][16 * 260];
  __shared__ __align__(16) float Kst[4][16 * 260];
  __shared__ float qbuf[4][DIM];
  __shared__ float attnw[4][NHEADS * MNB];

  const int w = threadIdx.x >> 5;
  const int lane = threadIdx.x & 31;
  const int n = blockIdx.x * 4 + w;

  // gather neighbor rows (zero-fill invalid)
  for (int m = 0; m < MNB; ++m) {
    int idx = nbr[n * MNB + m];
    float4* dst = (float4*)&Anf[w][m * 260];
    if (idx >= 0) {
      const float4* s4 = (const float4*)(src + (size_t)idx * DIM);
      for (int t = lane; t < DIM / 4; t += 32) dst[t] = s4[t];
    } else {
      float4 z = make_float4(0.f, 0.f, 0.f, 0.f);
      for (int t = lane; t < DIM / 4; t += 32) dst[t] = z;
    }
  }
  __syncthreads();

  // Q = nf@wq + bq (only row 0 is the query)
  for (int jt = 0; jt < 16; ++jt) {
    v8f c = {};
    c = gemm_tile_p(Anf[w], 260, p_wq, DIM / 4, jt, c);
    if (lane < 16) {
      int d = jt * 16 + lane;
      qbuf[w][d] = c[0] + bq[d];
    }
  }
  // K = nf@wk + bk, stored to LDS
  for (int jt = 0; jt < 16; ++jt) {
    v8f c = {};
    c = gemm_tile_p(Anf[w], 260, p_wk, DIM / 4, jt, c);
    int col = lane & 15, half = lane >> 4;
#pragma unroll
    for (int i = 0; i < 8; ++i) {
      int r = half * 8 + i, d = jt * 16 + col;
      Kst[w][r * 260 + d] = c[i] + bk[d];
    }
  }
  __syncthreads();

  // scores + softmax: lane -> (h in lane-half, m = lane&15); width-16 shuffles
  {
    int m = lane & 15;
    int half = lane >> 4;
    const float scale = 0.17677669529663687f;  // 1/sqrt(32)
    for (int hh = 0; hh < 4; ++hh) {
      int h = hh * 2 + half;
      float s = 0.f;
      const float* qp = &qbuf[w][h * DHEAD];
      const float* kp = &Kst[w][m * 260 + h * DHEAD];
#pragma unroll
      for (int d = 0; d < DHEAD; ++d) s += qp[d] * kp[d];
      s *= scale;
      float mx = s;
      for (int o = 1; o < 16; o <<= 1) mx = fmaxf(mx, __shfl_xor(mx, o, 32));
      float e = __expf(s - mx);
      float sum = e;
      for (int o = 1; o < 16; o <<= 1) sum += __shfl_xor(sum, o, 32);
      attnw[w][h * MNB + m] = e / sum;
    }
  }
  __syncthreads();

  // V tiles consumed directly into attention-weighted sum.
  for (int jt = 0; jt < 16; ++jt) {
    v8f c = {};
    c = gemm_tile_p(Anf[w], 260, p_wv, DIM / 4, jt, c);
    int col = lane & 15, half = lane >> 4;
    int h = jt >> 1;
    float partial = 0.f;
#pragma unroll
    for (int i = 0; i < 8; ++i) {
      int m = half * 8 + i;
      partial += attnw[w][h * MNB + m] * c[i];
    }
    partial += __shfl_xor(partial, 16, 32);
    if (lane < 16) {
      int d = jt * 16 + col;
      aog[(size_t)n * DIM + d] = partial + bv[d];
    }
  }
}

// ---------------- Stage D: wo + FFN + LN + fusion ----------------
__global__ __launch_bounds__(256) void post_kernel(
    const float* __restrict__ aog, const float* __restrict__ feats,
    const float* __restrict__ p_wo, const float* __restrict__ bo,
    const float* __restrict__ p_fw1, const float* __restrict__ fb1,
    const float* __restrict__ p_fw2, const float* __restrict__ fb2,
    const float* __restrict__ lng, const float* __restrict__ lnb,
    const float* __restrict__ p_fusw1, const float* __restrict__ fusb1,
    const float* __restrict__ p_fusw2a, const float* __restrict__ p_fusw2b,
    const float* __restrict__ fusb2, float* __restrict__ fout) {
  __shared__ __align__(16) float Ain[16 * 260];
  __shared__ __align__(16) float Obuf[16 * 260];
  __shared__ __align__(16) float Hbuf[16 * 516];
  __shared__ __align__(16) float Ybuf[16 * 260];
  __shared__ __align__(16) float Ft[16 * 68];
  __shared__ __align__(16) float Tt[16 * 68];

  const int tid = threadIdx.x;
  const int w = tid >> 5;
  const int lane = tid & 31;
  const int n0 = blockIdx.x * 16;

  for (int t = tid; t < 16 * 64; t += 256) {
    int row = t >> 6, qq = t & 63;
    ((float4*)&Ain[row * 260])[qq] =
        ((const float4*)(aog + (size_t)(n0 + row) * DIM))[qq];
  }
  for (int t = tid; t < 16 * 16; t += 256) {
    int row = t >> 4, qq = t & 15;
    ((float4*)&Ft[row * 68])[qq] =
        ((const float4*)(feats + (size_t)(n0 + row) * CIN))[qq];
  }
  __syncthreads();

  // O = AO@wo + bo
  for (int jj = 0; jj < 2; ++jj) {
    int jt = w * 2 + jj;
    v8f c = {};
    c = gemm_tile_p(Ain, 260, p_wo, DIM / 4, jt, c);
    int col = lane & 15, half = lane >> 4;
#pragma unroll
    for (int i = 0; i < 8; ++i) {
      int r = half * 8 + i, d = jt * 16 + col;
      Obuf[r * 260 + d] = c[i] + bo[d];
    }
  }
  __syncthreads();
  // H = relu(O@ffn_w1 + b1)
  for (int jj = 0; jj < 4; ++jj) {
    int jt = w * 4 + jj;
    v8f c = {};
    c = gemm_tile_p(Obuf, 260, p_fw1, DIM / 4, jt, c);
    int col = lane & 15, half = lane >> 4;
#pragma unroll
    for (int i = 0; i < 8; ++i) {
      int r = half * 8 + i, d = jt * 16 + col;
      Hbuf[r * 516 + d] = fmaxf(c[i] + fb1[d], 0.f);
    }
  }
  __syncthreads();
  // Z = O + H@ffn_w2 + b2
  for (int jj = 0; jj < 2; ++jj) {
    int jt = w * 2 + jj;
    v8f c = {};
    c = gemm_tile_p(Hbuf, 516, p_fw2, DFFNV / 4, jt, c);
    int col = lane & 15, half = lane >> 4;
#pragma unroll
    for (int i = 0; i < 8; ++i) {
      int r = half * 8 + i, d = jt * 16 + col;
      Ybuf[r * 260 + d] = c[i] + fb2[d] + Obuf[r * 260 + d];
    }
  }
  __syncthreads();
  // row-wise layernorm (2 rows per wave, wave32 shuffle reduction)
  for (int rr = 0; rr < 2; ++rr) {
    int r = w * 2 + rr;
    float s = 0.f;
    for (int t = lane; t < DIM; t += 32) s += Ybuf[r * 260 + t];
    for (int o = 1; o < 32; o <<= 1) s += __shfl_xor(s, o, 32);
    float mean = s / (float)DIM;
    float v = 0.f;
    for (int t = lane; t < DIM; t += 32) {
      float dd = Ybuf[r * 260 + t] - mean;
      v += dd * dd;
    }
    for (int o = 1; o < 32; o <<= 1) v += __shfl_xor(v, o, 32);
    float inv = rsqrtf(v / (float)DIM + EPSV);
    for (int t = lane; t < DIM; t += 32)
      Ybuf[r * 260 + t] = (Ybuf[r * 260 + t] - mean) * inv * lng[t] + lnb[t];
  }
  __syncthreads();
  // T = Y@fus_w1 + fus_b1
  if (w < 4) {
    int jt = w;
    v8f c = {};
    c = gemm_tile_p(Ybuf, 260, p_fusw1, DIM / 4, jt, c);
    int col = lane & 15, half = lane >> 4;
#pragma unroll
    for (int i = 0; i < 8; ++i) {
      int r = half * 8 + i, d = jt * 16 + col;
      Tt[r * 68 + d] = c[i] + fusb1[d];
    }
  }
  __syncthreads();
  // F = feats@fus_w2[0:64] + T@fus_w2[64:128] + fus_b2
  if (w < 4) {
    int jt = w;
    v8f c = {};
    c = gemm_tile_p(Ft, 68, p_fusw2a, CIN / 4, jt, c);
    c = gemm_tile_p(Tt, 68, p_fusw2b, CIN / 4, jt, c);
    int col = lane & 15, half = lane >> 4;
#pragma unroll
    for (int i = 0; i < 8; ++i) {
      int r = half * 8 + i, d = jt * 16 + col;
      fout[(size_t)(n0 + r) * CIN + d] = c[i] + fusb2[d];
    }
  }
}

// ---------------- Stage E: deterministic batchnorm ----------------
__global__ __launch_bounds__(256) void bn_partial_kernel(
    const float* __restrict__ f, float* __restrict__ part) {
  __shared__ float ls[256], lq[256];
  int tid = threadIdx.x;
  int ch = tid & 63, grp = tid >> 6;
  int base = blockIdx.x * 256 + grp * 64;
  float s = 0.f, sq = 0.f;
  for (int r = 0; r < 64; ++r) {
    float v = f[(size_t)(base + r) * CIN + ch];
    s += v;
    sq += v * v;
  }
  ls[tid] = s;
  lq[tid] = sq;
  __syncthreads();
  if (grp == 0) {
    float ts = ls[ch] + ls[64 + ch] + ls[128 + ch] + ls[192 + ch];
    float tq = lq[ch] + lq[64 + ch] + lq[128 + ch] + lq[192 + ch];
    part[blockIdx.x * 128 + ch] = ts;
    part[blockIdx.x * 128 + 64 + ch] = tq;
  }
}

__global__ void bn_final_kernel(const float* __restrict__ part,
                                float* __restrict__ stats) {
  int ch = threadIdx.x;  // 64 threads
  float s = 0.f, sq = 0.f;
  for (int b = 0; b < 64; ++b) {
    s += part[b * 128 + ch];
    sq += part[b * 128 + 64 + ch];
  }
  float mean = s / (float)NPTS;
  float var = sq / (float)NPTS - mean * mean;
  stats[ch] = mean;
  stats[64 + ch] = rsqrtf(var + EPSV);
}

__global__ void bn_apply_kernel(const float* __restrict__ f,
                                const float* __restrict__ stats,
                                const float* __restrict__ g,
                                const float* __restrict__ b,
                                float* __restrict__ out) {
  int i = blockIdx.x * blockDim.x + threadIdx.x;
  if (i >= NPTS * CIN) return;
  int ch = i & 63;
  float v = (f[i] - stats[ch]) * stats[64 + ch] * g[ch] + b[ch];
  out[i] = fmaxf(v, 0.f);
}

extern "C" void kernel_launch(void* const* d_in, const int* in_sizes, int n_in,
                              void* d_out, int out_size, void* d_ws,
                              size_t ws_size, hipStream_t stream) {
  (void)in_sizes; (void)n_in; (void)out_size; (void)ws_size;
  const int* coords = (const int*)d_in[0];
  const float* feats = (const float*)d_in[1];
  const float* pe_w1 = (const float*)d_in[2];
  const float* pe_b1 = (const float*)d_in[3];
  const float* pe_w2 = (const float*)d_in[4];
  const float* pe_b2 = (const float*)d_in[5];
  const float* proj_w = (const float*)d_in[6];
  const float* proj_b = (const float*)d_in[7];
  const float* wq = (const float*)d_in[8];
  const float* bq = (const float*)d_in[9];
  const float* wk = (const float*)d_in[10];
  const float* bk = (const float*)d_in[11];
  const float* wv = (const float*)d_in[12];
  const float* bv = (const float*)d_in[13];
  const float* wo = (const float*)d_in[14];
  const float* bo = (const float*)d_in[15];
  const float* ffn_w1 = (const float*)d_in[16];
  const float* ffn_b1 = (const float*)d_in[17];
  const float* ffn_w2 = (const float*)d_in[18];
  const float* ffn_b2 = (const float*)d_in[19];
  const float* ln_g = (const float*)d_in[20];
  const float* ln_b = (const float*)d_in[21];
  const float* fus_w1 = (const float*)d_in[22];
  const float* fus_b1 = (const float*)d_in[23];
  const float* fus_w2 = (const float*)d_in[24];
  const float* fus_b2 = (const float*)d_in[25];
  const float* bn_g = (const float*)d_in[26];
  const float* bn_b = (const float*)d_in[27];

  char* ws = (char*)d_ws;
  size_t off = 0;
  float* srcb = (float*)(ws + off); off += (size_t)NPTS * DIM * 4;
  float* hid  = (float*)(ws + off); off += (size_t)NPTS * 128 * 4;
  float* aob  = (float*)(ws + off); off += (size_t)NPTS * DIM * 4;
  float* fpre = (float*)(ws + off); off += (size_t)NPTS * CIN * 4;
  float* part = (float*)(ws + off); off += 64 * 128 * 4;
  float* stats = (float*)(ws + off); off += 128 * 4;
  int* nbr = (int*)(ws + off); off += (size_t)NPTS * MNB * 4;
  // fragment-major packed weights
  float* p_projw = (float*)(ws + off); off += (size_t)CIN * DIM * 4;
  float* p_pew2 = (float*)(ws + off); off += (size_t)128 * DIM * 4;
  float* p_wq = (float*)(ws + off); off += (size_t)DIM * DIM * 4;
  float* p_wk = (float*)(ws + off); off += (size_t)DIM * DIM * 4;
  float* p_wv = (float*)(ws + off); off += (size_t)DIM * DIM * 4;
  float* p_wo = (float*)(ws + off); off += (size_t)DIM * DIM * 4;
  float* p_fw1 = (float*)(ws + off); off += (size_t)DIM * DFFNV * 4;
  float* p_fw2 = (float*)(ws + off); off += (size_t)DFFNV * DIM * 4;
  float* p_fusw1 = (float*)(ws + off); off += (size_t)DIM * CIN * 4;
  float* p_fusw2a = (float*)(ws + off); off += (size_t)CIN * CIN * 4;
  float* p_fusw2b = (float*)(ws + off); off += (size_t)CIN * CIN * 4;

#define PACK(Wsrc, Kk, Oo, Dst)                                           \
  pack_b_kernel<<<((Kk) * (Oo) + 255) / 256, 256, 0, stream>>>(Wsrc, Kk, \
                                                               Oo, Dst)
  PACK(proj_w, CIN, DIM, p_projw);
  PACK(pe_w2, 128, DIM, p_pew2);
  PACK(wq, DIM, DIM, p_wq);
  PACK(wk, DIM, DIM, p_wk);
  PACK(wv, DIM, DIM, p_wv);
  PACK(wo, DIM, DIM, p_wo);
  PACK(ffn_w1, DIM, DFFNV, p_fw1);
  PACK(ffn_w2, DFFNV, DIM, p_fw2);
  PACK(fus_w1, DIM, CIN, p_fusw1);
  PACK(fus_w2, CIN, CIN, p_fusw2a);
  PACK(fus_w2 + CIN * CIN, CIN, CIN, p_fusw2b);
#undef PACK

  pe_hidden_kernel<<<(NPTS * 128) / 256, 256, 0, stream>>>(coords, pe_w1,
                                                           pe_b1, hid);
  neighbor_kernel<<<NPTS / 256, 256, 0, stream>>>(coords, nbr);
  src_kernel<<<NPTS / 16, 256, 0, stream>>>(feats, hid, p_projw, proj_b,
                                            p_pew2, pe_b2, srcb);
  attn_kernel<<<NPTS / 4, 128, 0, stream>>>(srcb, nbr, p_wq, bq, p_wk, bk,
                                            p_wv, bv, aob);
  post_kernel<<<NPTS / 16, 256, 0, stream>>>(aob, feats, p_wo, bo, p_fw1,
                                             ffn_b1, p_fw2, ffn_b2, ln_g, ln_b,
                                             p_fusw1, fus_b1, p_fusw2a,
                                             p_fusw2b, fus_b2, fpre);
  bn_partial_kernel<<<64, 256, 0, stream>>>(fpre, part);
  bn_final_kernel<<<1, 64, 0, stream>>>(part, stats);
  bn_apply_kernel<<<(NPTS * CIN) / 256, 256, 0, stream>>>(fpre, stats, bn_g,
                                                          bn_b, (float*)d_out);
}